// SVDBlock_22565758174140
// MI455X (gfx1250) — compile-verified
//
#include <hip/hip_runtime.h>
#include <hip/hip_bf16.h>
#include <math.h>

typedef __attribute__((ext_vector_type(2))) float v2f;
typedef __attribute__((ext_vector_type(8))) float v8f;

#define NPTS 4096
#define N4   (NPTS / 4)

// ---------------------------------------------------------------------------
// Kernel 1: per-batch weighted statistics T[4][4], T[i][j] = sum_n w*Ki*Pj
// (K3 = P3 = 1). One block (256 threads = 8 waves) per batch. Streaming
// phase is b128-coalesced FMA (memory bound, ~235MB @ 23.3TB/s ~ 10us).
// Final 16x256 cross-thread reduction done with V_WMMA_F32_16X16X4_F32:
//   C(16x16) += A(16x4) * ones(4x16), 64 WMMAs -> row sums of partials.
// ---------------------------------------------------------------------------
__device__ __forceinline__ void upd(float acc[16], float wv,
                                    float k0, float k1, float k2,
                                    float p0, float p1, float p2) {
    float wk0 = wv * k0, wk1 = wv * k1, wk2 = wv * k2;
    acc[0]  += wk0 * p0;  acc[1]  += wk0 * p1;  acc[2]  += wk0 * p2;  acc[3]  += wk0;
    acc[4]  += wk1 * p0;  acc[5]  += wk1 * p1;  acc[6]  += wk1 * p2;  acc[7]  += wk1;
    acc[8]  += wk2 * p0;  acc[9]  += wk2 * p1;  acc[10] += wk2 * p2;  acc[11] += wk2;
    acc[12] += wv  * p0;  acc[13] += wv  * p1;  acc[14] += wv  * p2;  acc[15] += wv;
}

__global__ __launch_bounds__(256) void stats_kernel(const float* __restrict__ kp,
                                                    const float* __restrict__ ps,
                                                    const float* __restrict__ w,
                                                    float* __restrict__ stats) {
    const int b = blockIdx.x;
    const int t = threadIdx.x;

    const float4* kp0 = (const float4*)(kp + (size_t)b * 3 * NPTS);
    const float4* kp1 = kp0 + N4;
    const float4* kp2 = kp1 + N4;
    const float4* ps0 = (const float4*)(ps + (size_t)b * 3 * NPTS);
    const float4* ps1 = ps0 + N4;
    const float4* ps2 = ps1 + N4;
    const float4* wp  = (const float4*)(w + (size_t)b * NPTS);

    float acc[16];
#pragma unroll
    for (int i = 0; i < 16; ++i) acc[i] = 0.0f;

#pragma unroll
    for (int it = 0; it < N4 / 256; ++it) {
        int j = it * 256 + t;
        float4 w4 = wp[j];
        float4 a0 = kp0[j], a1 = kp1[j], a2 = kp2[j];
        float4 b0 = ps0[j], b1 = ps1[j], b2 = ps2[j];
        upd(acc, w4.x, a0.x, a1.x, a2.x, b0.x, b1.x, b2.x);
        upd(acc, w4.y, a0.y, a1.y, a2.y, b0.y, b1.y, b2.y);
        upd(acc, w4.z, a0.z, a1.z, a2.z, b0.z, b1.z, b2.z);
        upd(acc, w4.w, a0.w, a1.w, a2.w, b0.w, b1.w, b2.w);
    }

    // red[stat][thread], padded to keep LDS banks conflict-free both ways
    __shared__ float red[16][257];
#pragma unroll
    for (int a = 0; a < 16; ++a) red[a][t] = acc[a];
    __syncthreads();

    // Wave 0 (lanes 0..31, EXEC all ones) reduces 16x256 via WMMA f32 16x16x4.
    if (t < 32) {
        v8f c = {0.f, 0.f, 0.f, 0.f, 0.f, 0.f, 0.f, 0.f};
        v2f ones;
        ones.x = 1.0f; ones.y = 1.0f;
        const int l15 = t & 15;
        const int hi  = t >> 4;
        // A(16x4) layout: lane L: V0=A[L][0],V1=A[L][1]; lane L+16: V0=A[L][2],V1=A[L][3]
#pragma unroll 8
        for (int i = 0; i < 64; ++i) {
            int t0 = 4 * i + 2 * hi;
            v2f a;
            a.x = red[l15][t0];
            a.y = red[l15][t0 + 1];
            // C[m][n] += sum_k A[m][k] * 1  -> sums 4 threads' partials per stat m
            c = __builtin_amdgcn_wmma_f32_16x16x4_f32(
                    false, a, false, ones, (short)0, c, false, false);
        }
        // C layout: VGPR r: lanes 0-15 -> M=r, lanes 16-31 -> M=r+8; take N=0 column
        if (t == 0) {
#pragma unroll
            for (int r = 0; r < 8; ++r) stats[(size_t)b * 16 + r] = c[r];
        }
        if (t == 16) {
#pragma unroll
            for (int r = 0; r < 8; ++r) stats[(size_t)b * 16 + 8 + r] = c[r];
        }
    }
}

// ---------------------------------------------------------------------------
// Kernel 2: one thread per batch. Horn quaternion solve (== SVD Kabsch with
// reflection fix): dominant eigenvector of 4x4 symmetric N via shifted
// matrix squaring (power 2^16), then R from quaternion, t = pc - R*kc.
// ---------------------------------------------------------------------------
__global__ __launch_bounds__(256) void solve_kernel(const float* __restrict__ stats,
                                                    float* __restrict__ out, int NB) {
    int b = blockIdx.x * blockDim.x + threadIdx.x;
    if (b >= NB) return;

    float T[16];
#pragma unroll
    for (int i = 0; i < 16; ++i) T[i] = stats[(size_t)b * 16 + i];

    const float W    = T[15];
    const float invW = 1.0f / W;
    float kc[3] = {T[3] * invW, T[7] * invW, T[11] * invW};
    float pc[3] = {T[12] * invW, T[13] * invW, T[14] * invW};

    float H[3][3];
#pragma unroll
    for (int i = 0; i < 3; ++i)
#pragma unroll
        for (int j = 0; j < 3; ++j)
            H[i][j] = T[i * 4 + j] * invW - kc[i] * pc[j];

    const float Sxx = H[0][0], Sxy = H[0][1], Sxz = H[0][2];
    const float Syx = H[1][0], Syy = H[1][1], Syz = H[1][2];
    const float Szx = H[2][0], Szy = H[2][1], Szz = H[2][2];

    float Nm[4][4];
    Nm[0][0] =  Sxx + Syy + Szz;
    Nm[0][1] =  Syz - Szy;  Nm[0][2] = Szx - Sxz;  Nm[0][3] = Sxy - Syx;
    Nm[1][1] =  Sxx - Syy - Szz;
    Nm[1][2] =  Sxy + Syx;  Nm[1][3] = Szx + Sxz;
    Nm[2][2] = -Sxx + Syy - Szz;
    Nm[2][3] =  Syz + Szy;
    Nm[3][3] = -Sxx - Syy + Szz;
    Nm[1][0] = Nm[0][1]; Nm[2][0] = Nm[0][2]; Nm[3][0] = Nm[0][3];
    Nm[2][1] = Nm[1][2]; Nm[3][1] = Nm[1][3]; Nm[3][2] = Nm[2][3];

    // Shift so dominant eigenvalue of (N + sigma I) is the max-|.| one.
    float fr = 0.0f;
#pragma unroll
    for (int i = 0; i < 4; ++i)
#pragma unroll
        for (int j = 0; j < 4; ++j) fr += Nm[i][j] * Nm[i][j];
    fr = sqrtf(fr) + 1e-20f;
#pragma unroll
    for (int d = 0; d < 4; ++d) Nm[d][d] += fr;

    // 16 rounds of trace-normalized squaring -> projector onto dominant eigvec.
    float Bm[4][4];
#pragma unroll
    for (int i = 0; i < 4; ++i)
#pragma unroll
        for (int j = 0; j < 4; ++j) Bm[i][j] = Nm[i][j];

    for (int it = 0; it < 16; ++it) {
        float tr = Bm[0][0] + Bm[1][1] + Bm[2][2] + Bm[3][3] + 1e-30f;
        float s  = 1.0f / tr;
        float Sc[4][4], Sq[4][4];
#pragma unroll
        for (int i = 0; i < 4; ++i)
#pragma unroll
            for (int j = 0; j < 4; ++j) Sc[i][j] = Bm[i][j] * s;
#pragma unroll
        for (int i = 0; i < 4; ++i)
#pragma unroll
            for (int j = 0; j < 4; ++j) {
                float acc = 0.0f;
#pragma unroll
                for (int k = 0; k < 4; ++k) acc += Sc[i][k] * Sc[k][j];
                Sq[i][j] = acc;
            }
#pragma unroll
        for (int i = 0; i < 4; ++i)
#pragma unroll
            for (int j = 0; j < 4; ++j) Bm[i][j] = Sq[i][j];
    }

    // Dominant eigenvector = column with largest diagonal of the projector.
    int cmax = 0;
    float dmax = Bm[0][0];
    if (Bm[1][1] > dmax) { dmax = Bm[1][1]; cmax = 1; }
    if (Bm[2][2] > dmax) { dmax = Bm[2][2]; cmax = 2; }
    if (Bm[3][3] > dmax) { dmax = Bm[3][3]; cmax = 3; }
    float q[4];
#pragma unroll
    for (int i = 0; i < 4; ++i)
        q[i] = (cmax == 0) ? Bm[i][0] : (cmax == 1) ? Bm[i][1]
             : (cmax == 2) ? Bm[i][2] : Bm[i][3];
    float qn = rsqrtf(q[0]*q[0] + q[1]*q[1] + q[2]*q[2] + q[3]*q[3] + 1e-30f);
    const float q0 = q[0]*qn, qx = q[1]*qn, qy = q[2]*qn, qz = q[3]*qn;

    float R[3][3];
    R[0][0] = 1.0f - 2.0f*(qy*qy + qz*qz);
    R[0][1] = 2.0f*(qx*qy - qz*q0);
    R[0][2] = 2.0f*(qx*qz + qy*q0);
    R[1][0] = 2.0f*(qx*qy + qz*q0);
    R[1][1] = 1.0f - 2.0f*(qx*qx + qz*qz);
    R[1][2] = 2.0f*(qy*qz - qx*q0);
    R[2][0] = 2.0f*(qx*qz - qy*q0);
    R[2][1] = 2.0f*(qy*qz + qx*q0);
    R[2][2] = 1.0f - 2.0f*(qx*qx + qy*qy);

#pragma unroll
    for (int i = 0; i < 3; ++i)
#pragma unroll
        for (int j = 0; j < 3; ++j)
            out[(size_t)b * 9 + i * 3 + j] = R[i][j];

#pragma unroll
    for (int i = 0; i < 3; ++i) {
        float ti = pc[i] - (R[i][0]*kc[0] + R[i][1]*kc[1] + R[i][2]*kc[2]);
        out[(size_t)NB * 9 + (size_t)b * 3 + i] = ti;
    }
}

// ---------------------------------------------------------------------------
extern "C" void kernel_launch(void* const* d_in, const int* in_sizes, int n_in,
                              void* d_out, int out_size, void* d_ws, size_t ws_size,
                              hipStream_t stream) {
    const float* kp = (const float*)d_in[0];   // [B,3,N]
    const float* ps = (const float*)d_in[1];   // [B,3,N]
    const float* w  = (const float*)d_in[2];   // [B,1,N]
    float* out   = (float*)d_out;              // [B*9] R then [B*3] t
    float* stats = (float*)d_ws;               // [B,16]

    const int NB = in_sizes[2] / NPTS;         // B = 2048

    stats_kernel<<<NB, 256, 0, stream>>>(kp, ps, w, stats);
    solve_kernel<<<(NB + 255) / 256, 256, 0, stream>>>(stats, out, NB);
}